// MessagePassing_49744311222856
// MI455X (gfx1250) — compile-verified
//
#include <hip/hip_runtime.h>

// out = segment_sum(x[col], row)   with x: [N,64] f32, edge_index: [2,E] int
//
// Memory-bound irregular gather/scatter. Entire working set (~61 MB) fits in
// MI455X's 192 MB L2, so the targeted hardware paths are:
//   - 128-bit gathers (global_load_b128): 16 lanes per edge * float4
//   - no-return f32 atomics (global_atomic_add_f32, STOREcnt, resolved at L2)
//   - prefetch of the streamed edge list (global_prefetch_b8)
// WMMA is inapplicable: random destination rows cannot be tiled into dense
// 16x16 accumulator windows, so matrix units would only add traffic.
//
// This kernel is placed first in the file so its body shows in the disasm
// snippet (verifying native atomics / b128 gathers / prefetch).

#define D_FEAT 64

__device__ __forceinline__ void process_edge(const float* __restrict__ x,
                                             const int*   __restrict__ edge_row,
                                             const int*   __restrict__ edge_col,
                                             float*       __restrict__ out,
                                             int e, int lane4) {
    const int row = edge_row[e];
    const int col = edge_col[e];

    // 128-bit gather of this lane's 4-float (1/16) slice of the row
    const float4 v =
        *reinterpret_cast<const float4*>(x + (size_t)col * D_FEAT + lane4 * 4);

    float* dst = out + (size_t)row * D_FEAT + lane4 * 4;
    // hardware f32 atomics, no return value -> global_atomic_add_f32
    unsafeAtomicAdd(dst + 0, v.x);
    unsafeAtomicAdd(dst + 1, v.y);
    unsafeAtomicAdd(dst + 2, v.z);
    unsafeAtomicAdd(dst + 3, v.w);
}

__global__ __launch_bounds__(256)
void scatter_add_kernel(const float* __restrict__ x,
                        const int*   __restrict__ edge_row,
                        const int*   __restrict__ edge_col,
                        float*       __restrict__ out,
                        int n_edges) {
    const int gtid   = blockIdx.x * blockDim.x + threadIdx.x;
    const int lane4  = gtid & 15;               // 16 lanes per edge, float4 each
    int       e      = gtid >> 4;
    const int estrid = (gridDim.x * blockDim.x) >> 4;

    // unroll x2: two independent gathers in flight per iteration (MLP)
    for (; e + estrid < n_edges; e += 2 * estrid) {
        const int ep = e + 2 * estrid;
        if (ep < n_edges) {                     // stream-ahead -> global_prefetch_b8
            __builtin_prefetch(&edge_row[ep], 0, 3);
            __builtin_prefetch(&edge_col[ep], 0, 3);
        }
        process_edge(x, edge_row, edge_col, out, e,          lane4);
        process_edge(x, edge_row, edge_col, out, e + estrid, lane4);
    }
    if (e < n_edges) {
        process_edge(x, edge_row, edge_col, out, e, lane4);
    }
}

__global__ void zero_out_kernel(float4* __restrict__ out4, int n4) {
    int i = blockIdx.x * blockDim.x + threadIdx.x;
    if (i < n4) out4[i] = make_float4(0.f, 0.f, 0.f, 0.f);
}

extern "C" void kernel_launch(void* const* d_in, const int* in_sizes, int n_in,
                              void* d_out, int out_size, void* d_ws, size_t ws_size,
                              hipStream_t stream) {
    const float* x   = (const float*)d_in[0];
    const int*   ei  = (const int*)d_in[1];   // [2, E] flat: first E = row, next E = col
    float*       out = (float*)d_out;

    const int n_edges = in_sizes[1] / 2;

    // 1) zero the (poisoned) output: out_size floats, vectorized
    const int n4 = out_size / 4;
    zero_out_kernel<<<(n4 + 255) / 256, 256, 0, stream>>>((float4*)out, n4);

    // 2) scatter-add: persistent grid-stride, 16 lanes/edge
    //    4096 blocks * 256 thr = 1M threads = 64K edge slots (~19 iterations)
    const int blocks = 4096;
    scatter_add_kernel<<<blocks, 256, 0, stream>>>(x, ei, ei + n_edges, out, n_edges);
}